// SectionTwo_11381663334663
// MI455X (gfx1250) — compile-verified
//
#include <hip/hip_runtime.h>
#include <stdint.h>

// Problem constants from the reference: HEIGHT=512, WIDTH=960, MAXDISP=32
#define H2 256   // HEIGHT/2
#define W2 480   // WIDTH/2
#define DMAX 32  // MAXDISP
#define CCH 32   // channels per image
#define W4 (W2 / 4)  // 120 float4 per row

typedef float v4f __attribute__((ext_vector_type(4)));
typedef int   v4i_vs __attribute__((vector_size(16)));  // matches builtin param type

// out layout: [c2 (64)][d (32)][h (256)][w (480)], float32.
//   c2 <  32: out = imgl[c2][h][w]                       (broadcast over d)
//   c2 >= 32: out = (w >= d) ? imgr[c2-32][h][w-d] : 0   (shifted, zero-filled)
//
// Store-bound kernel (~1 GB written once). Non-temporal b128 stores keep the
// streaming output from evicting the fully L2-resident inputs. Right half
// stages the imgr row into LDS (with a 32-float zero apron) via the gfx1250
// async-to-LDS path (ASYNCcnt), making the shifted read branch-free.
__global__ __launch_bounds__(128) void costvol_kernel(
    const float* __restrict__ imgl,
    const float* __restrict__ imgr,
    float* __restrict__ out)
{
    const int tid = threadIdx.x;
    const int h   = blockIdx.x;   // 0..255
    const int d   = blockIdx.y;   // 0..31
    const int c2  = blockIdx.z;   // 0..63

    float* __restrict__ orow =
        out + (((size_t)c2 * DMAX + d) * H2 + h) * W2;

    // ---------------- left half: pure broadcast copy ----------------
    if (c2 < CCH) {
        if (tid < W4) {
            const v4f* __restrict__ src =
                (const v4f*)(imgl + ((size_t)c2 * H2 + h) * W2);
            v4f v = src[tid];                       // L2-resident, reused 32x over d
            __builtin_nontemporal_store(v, (v4f*)orow + tid);  // stream out, TH=NT
        }
        return;
    }

    // ---------------- right half: shifted gather with zero fill ----------------
    const int c = c2 - CCH;
    const float* __restrict__ src = imgr + ((size_t)c * H2 + h) * W2;

    // Zero apron of DMAX floats in front => srow[DMAX + (w - d)] is always in
    // range for w in [0,480), d in [0,32): index in [1, 511].
    __shared__ float srow[DMAX + W2];

    if (tid < DMAX) srow[tid] = 0.0f;

#if __has_builtin(__builtin_amdgcn_global_load_async_to_lds_b128)
    // gfx1250 async copy: global -> LDS without touching VGPRs (ASYNCcnt).
    // Builtin takes generic (flat) pointers to 16-byte vectors; backend
    // resolves the global/LDS address spaces.
    if (tid < W4) {
        v4i_vs* gsrc = (v4i_vs*)src + tid;                 // points into global
        v4i_vs* ldst = (v4i_vs*)(srow + DMAX) + tid;       // points into LDS
        __builtin_amdgcn_global_load_async_to_lds_b128(gsrc, ldst, 0, 0);
    }
#if __has_builtin(__builtin_amdgcn_s_wait_asynccnt)
    __builtin_amdgcn_s_wait_asynccnt(0);
#else
    asm volatile("s_wait_asynccnt 0" ::: "memory");
#endif
#else
    // Fallback: plain vectorized staging through VGPRs (rows are 16B aligned).
    if (tid < W4) {
        ((v4f*)(srow + DMAX))[tid] = ((const v4f*)src)[tid];
    }
#endif

    __syncthreads();

    if (tid < W4) {
        const int w0 = tid * 4;
        const float* p = srow + (DMAX - d) + w0;  // = srow[DMAX + (w - d)]
        v4f v;
        v.x = p[0];
        v.y = p[1];
        v.z = p[2];
        v.w = p[3];
        __builtin_nontemporal_store(v, (v4f*)orow + tid);
    }
}

extern "C" void kernel_launch(void* const* d_in, const int* in_sizes, int n_in,
                              void* d_out, int out_size, void* d_ws, size_t ws_size,
                              hipStream_t stream) {
    const float* imgl = (const float*)d_in[0];  // (1, 32, 256, 480) f32
    const float* imgr = (const float*)d_in[1];  // (1, 32, 256, 480) f32
    float* out = (float*)d_out;                 // (1, 64, 32, 256, 480) f32

    dim3 grid(H2, DMAX, 2 * CCH);  // (256, 32, 64): one block per output row
    dim3 block(128);               // 4 wave32; lanes 0..119 active (one float4 each)
    costvol_kernel<<<grid, block, 0, stream>>>(imgl, imgr, out);
}